// DistanceRestraint_29231547417073
// MI455X (gfx1250) — compile-verified
//
#include <hip/hip_runtime.h>
#include <hip/hip_bf16.h>
#include <math.h>

// Problem (reference): B=4, L=1024, P=262144, NSEG=36, NCUT=37.
// CB fits in LDS: 4*1024*3 floats = 48KB  (WGP LDS = 320KB).
#define MAX_CB_FLOATS 12288   // B*L*3
#define MAX_CUT        64     // >= NSEG+1

typedef float v4f __attribute__((ext_vector_type(4)));

__global__ __launch_bounds__(256)
void cbcb_pair_kernel(const float* __restrict__ CB,
                      const float* __restrict__ coeff,
                      const float* __restrict__ cutoffs,
                      const int*   __restrict__ pair_i,
                      const int*   __restrict__ pair_j,
                      float*       __restrict__ partial,
                      int P, int L, int B, int nseg, int ncut)
{
    __shared__ float s_cb[MAX_CB_FLOATS];
    __shared__ float s_cut[MAX_CUT];
    __shared__ float s_wred[8];

    const int tid = threadIdx.x;
    const int nthreads = blockDim.x;

    // ---- Stage CB into LDS with CDNA5 async global->LDS b128 copies ----
    const int ncb = B * L * 3;
    const int n4  = ncb >> 2;                 // number of 16B chunks
    for (int q = tid; q < n4; q += nthreads) {
        // Low 32 bits of the generic pointer to shared mem == LDS byte offset.
        unsigned int lds_off =
            (unsigned int)(unsigned long long)(&s_cb[0]) + (unsigned int)(q * 16);
        const float* gp = CB + (q << 2);
        asm volatile("global_load_async_to_lds_b128 %0, %1, off"
                     :: "v"(lds_off), "v"(gp)
                     : "memory");
    }
    // Scalar tail (none for the reference shapes, kept for generality).
    for (int q = (n4 << 2) + tid; q < ncb; q += nthreads) s_cb[q] = CB[q];
    // Cutoffs are tiny: plain load + ds store.
    if (tid < ncut && tid < MAX_CUT) s_cut[tid] = cutoffs[tid];

    // Wait for this wave's async transfers, then workgroup barrier so every
    // wave's LDS writes are visible to all.
    asm volatile("s_wait_asynccnt 0" ::: "memory");
    __syncthreads();

    const float cutmax = s_cut[ncut - 1];

    // ---- Main gather/eval loop: grid-stride over pairs ----
    float sum = 0.0f;
    const int gstride = gridDim.x * blockDim.x;
    for (int p = blockIdx.x * blockDim.x + tid; p < P; p += gstride) {
        const int i = pair_i[p];
        const int j = pair_j[p];
        const v4f* crow = (const v4f*)coeff +
                          ((size_t)i * (size_t)L + (size_t)j) * (size_t)nseg;
        #pragma unroll 4
        for (int b = 0; b < B; ++b) {
            const float* xp = &s_cb[(b * L + i) * 3];
            const float* yp = &s_cb[(b * L + j) * 3];
            const float dx = xp[0] - yp[0];
            const float dy = xp[1] - yp[1];
            const float dz = xp[2] - yp[2];
            const float d  = sqrtf(fmaf(dx, dx, fmaf(dy, dy, dz * dz)));
            if (d <= cutmax) {
                // searchsorted(cutoffs, d, side='left'): smallest k with cut[k] >= d
                int lo = 0, hi = ncut;
                while (lo < hi) {
                    const int m = (lo + hi) >> 1;
                    if (s_cut[m] < d) lo = m + 1; else hi = m;
                }
                int idx = lo - 1;
                idx = idx < 0 ? 0 : (idx > nseg - 1 ? nseg - 1 : idx);
                const float xr = d - s_cut[idx];
                // Random 16B gather into the 603MB coeff table: zero reuse, so
                // mark non-temporal (th:TH_LOAD_NT) to avoid thrashing L2.
                const v4f c = __builtin_nontemporal_load(crow + idx);
                sum += fmaf(fmaf(fmaf(c.x, xr, c.y), xr, c.z), xr, c.w);
            }
        }
    }

    // ---- Deterministic reduction: wave32 shuffles, then per-block LDS tree ----
    #pragma unroll
    for (int off = 16; off > 0; off >>= 1)
        sum += __shfl_down(sum, off, 32);
    const int wave = tid >> 5;
    const int lane = tid & 31;
    if (lane == 0) s_wred[wave] = sum;
    __syncthreads();
    if (tid == 0) {
        float t = 0.0f;
        const int nw = nthreads >> 5;
        for (int w = 0; w < nw; ++w) t += s_wred[w];
        partial[blockIdx.x] = t;
    }
}

__global__ __launch_bounds__(256)
void cbcb_reduce_kernel(const float* __restrict__ partial, int n,
                        float* __restrict__ out)
{
    __shared__ float s[256];
    float v = 0.0f;
    for (int i = threadIdx.x; i < n; i += 256) v += partial[i];
    s[threadIdx.x] = v;
    __syncthreads();
    #pragma unroll
    for (int off = 128; off > 0; off >>= 1) {
        if (threadIdx.x < off) s[threadIdx.x] += s[threadIdx.x + off];
        __syncthreads();
    }
    if (threadIdx.x == 0) out[0] = s[0];
}

extern "C" void kernel_launch(void* const* d_in, const int* in_sizes, int n_in,
                              void* d_out, int out_size, void* d_ws, size_t ws_size,
                              hipStream_t stream)
{
    (void)n_in; (void)out_size;
    const float* CB      = (const float*)d_in[0];
    const float* coeff   = (const float*)d_in[1];
    const float* cutoffs = (const float*)d_in[2];
    const int*   pair_i  = (const int*)d_in[3];
    const int*   pair_j  = (const int*)d_in[4];

    const int ncut = in_sizes[2];
    const int nseg = ncut - 1;
    const int P    = in_sizes[3];
    // coeff elems = L*L*nseg*4  ->  L
    const int L = (int)(0.5 + sqrt((double)in_sizes[1] / (4.0 * (double)nseg)));
    const int B = in_sizes[0] / (3 * L);

    const int threads = 256;                       // 8 wave32s per workgroup
    int blocks = (P + threads - 1) / threads;      // 1024 for reference shapes
    if (blocks > 1024) blocks = 1024;
    const size_t max_partials = ws_size / sizeof(float);
    if ((size_t)blocks > max_partials) blocks = (int)max_partials;
    if (blocks < 1) blocks = 1;

    float* partial = (float*)d_ws;

    cbcb_pair_kernel<<<blocks, threads, 0, stream>>>(
        CB, coeff, cutoffs, pair_i, pair_j, partial, P, L, B, nseg, ncut);
    cbcb_reduce_kernel<<<1, 256, 0, stream>>>(partial, blocks, (float*)d_out);
}